// VISION_32684701123090
// MI455X (gfx1250) — compile-verified
//
#include <hip/hip_runtime.h>
#include <hip/hip_bf16.h>

typedef float v2f __attribute__((ext_vector_type(2)));
typedef float v8f __attribute__((ext_vector_type(8)));

// Problem constants (from reference setup_inputs)
#define Bsz 8
#define Csz 64
#define Hsz 128
#define Wsz 128
#define HW  (Hsz * Wsz)          // 16384
#define Hp  130
#define Wp  130
#define HPWP (Hp * Wp)           // 16900
#define NTAP 9

// ---------------------------------------------------------------------------
// Kernel 1: zero-pad x  [B,C,128,128] -> [B,C,130,130]
// ---------------------------------------------------------------------------
__global__ void pad_kernel(const float* __restrict__ x, float* __restrict__ xpad, int total) {
    int t = blockIdx.x * blockDim.x + threadIdx.x;
    if (t >= total) return;
    int q  = t % HPWP;
    int bc = t / HPWP;
    int r  = q / Wp;
    int c  = q % Wp;
    float v = 0.0f;
    if (r >= 1 && r <= Hsz && c >= 1 && c <= Wsz)
        v = x[(size_t)bc * HW + (size_t)(r - 1) * Wsz + (c - 1)];
    xpad[t] = v;
}

// ---------------------------------------------------------------------------
// Kernel 2: reorder conv_w [co, c, kx, ky] -> Wr[n][c][co]  (n = kx*3+ky)
// ---------------------------------------------------------------------------
__global__ void reorder_w_kernel(const float* __restrict__ cw, float* __restrict__ Wr) {
    int t = blockIdx.x * blockDim.x + threadIdx.x;
    if (t >= NTAP * Csz * Csz) return;
    int co = t & 63;
    int c  = (t >> 6) & 63;
    int n  = t >> 12;
    Wr[t] = cw[((size_t)co * Csz + c) * NTAP + n];
}

// ---------------------------------------------------------------------------
// Kernel 3: offset conv (18 channels, 3x3, pad 1, bias). LDS-tiled direct conv.
// Grid: B*8*8 blocks of 256 threads; each block = one 16x16 spatial tile.
// ---------------------------------------------------------------------------
__global__ __launch_bounds__(256) void offset_conv_kernel(
    const float* __restrict__ x,       // [B,64,128,128]
    const float* __restrict__ pw,      // [18,64,3,3]
    const float* __restrict__ pb,      // [18]
    float* __restrict__ off)           // [B,18,128,128]
{
    __shared__ float xt[8 * 18 * 18];  // 8 channels, 18x18 halo tile (10.4 KB)
    __shared__ float wt[18 * 8 * 9];   // 18 oc x 8 c x 9 taps (5.2 KB)

    const int tid = threadIdx.x;
    const int tx = tid & 15;
    const int ty = tid >> 4;
    const int tilej = (blockIdx.x & 7) * 16;
    const int tilei = ((blockIdx.x >> 3) & 7) * 16;
    const int b = blockIdx.x >> 6;
    const int i = tilei + ty;
    const int j = tilej + tx;

    float acc[18];
    #pragma unroll
    for (int oc = 0; oc < 18; ++oc) acc[oc] = pb[oc];

    const float* xb = x + (size_t)b * Csz * HW;

    for (int cb = 0; cb < 8; ++cb) {
        __syncthreads();
        // stage x tile with halo (zero outside image)
        for (int t = tid; t < 8 * 18 * 18; t += 256) {
            int c  = t / 324;
            int rr = (t % 324) / 18;
            int cc = t % 18;
            int gi = tilei - 1 + rr;
            int gj = tilej - 1 + cc;
            float v = 0.0f;
            if (gi >= 0 && gi < Hsz && gj >= 0 && gj < Wsz)
                v = xb[(size_t)(cb * 8 + c) * HW + (size_t)gi * Wsz + gj];
            xt[t] = v;
        }
        // stage weight chunk
        for (int t = tid; t < 18 * 8 * 9; t += 256) {
            int oc  = t / 72;
            int c   = (t % 72) / 9;
            int tap = t % 9;
            wt[t] = pw[((size_t)oc * Csz + cb * 8 + c) * 9 + tap];
        }
        __syncthreads();

        for (int c = 0; c < 8; ++c) {
            #pragma unroll
            for (int tap = 0; tap < 9; ++tap) {
                float v = xt[c * 324 + (ty + tap / 3) * 18 + (tx + tap % 3)];
                #pragma unroll
                for (int oc = 0; oc < 18; ++oc)
                    acc[oc] = fmaf(wt[oc * 72 + c * 9 + tap], v, acc[oc]);
            }
        }
    }

    #pragma unroll
    for (int oc = 0; oc < 18; ++oc)
        off[(((size_t)b * 18 + oc) * Hsz + i) * Wsz + j] = acc[oc];
}

// ---------------------------------------------------------------------------
// Kernel 4: fused bilinear gather + GEMM via V_WMMA_F32_16X16X4_F32.
// GEMM view: [pixels=131072, K=576] x [K=576, 64] ; K ordered n*64 + c.
// One wave = 16 pixels x 64 out-channels (4 accumulator tiles).
// Block = 128 threads (4 waves, 4 consecutive M-tiles sharing the LDS B slab).
// ---------------------------------------------------------------------------
__global__ __launch_bounds__(128) void deform_gemm_kernel(
    const float* __restrict__ xpad,   // [B,64,130,130]
    const float* __restrict__ off,    // [B,18,128,128]
    const float* __restrict__ Wr,     // [9][64][64]  (n, c, co)
    float* __restrict__ out)          // [B,64,128,128]
{
    __shared__ float wchunk[64 * 64];            // 16 KB: tap-n weight slab [c][co]
    __shared__ float ostage[4][16 * 65];         // 16.6 KB: padded C transpose buffer

    const int wave = threadIdx.x >> 5;
    const int lane = threadIdx.x & 31;
    const int hl   = lane >> 4;       // which half of the wave (K pairing)
    const int nid  = lane & 15;       // matrix row (pixel) / col (co) id

    const int mtile = blockIdx.x * 4 + wave;     // 0..8191
    const int b     = mtile >> 10;               // 1024 M-tiles per image
    const int pix0  = (mtile & 1023) * 16;
    const int pix   = pix0 + nid;                // this lane's pixel (A row)
    const int i     = pix >> 7;                  // row in image
    const int j     = pix & 127;                 // col in image

    const float* xb = xpad + (size_t)b * Csz * HPWP;

    v8f acc0 = {}, acc1 = {}, acc2 = {}, acc3 = {};

    for (int n = 0; n < NTAP; ++n) {
        __syncthreads();  // protect previous slab reads
        {   // cooperative 16 KB weight slab load (float4)
            const float4* src = (const float4*)(Wr + (size_t)n * Csz * Csz);
            float4* dst = (float4*)wchunk;
            #pragma unroll
            for (int t = 0; t < 8; ++t)
                dst[threadIdx.x + t * 128] = src[threadIdx.x + t * 128];
        }
        __syncthreads();

        // --- per-(pixel, tap) sampling setup, reused across all 64 channels ---
        const float offx = off[(((size_t)b * 18 + n) * Hsz + i) * Wsz + j];
        const float offy = off[(((size_t)b * 18 + 9 + n) * Hsz + i) * Wsz + j];
        const float px = offx + (float)(i + 1) + (float)(n / 3 - 1);
        const float py = offy + (float)(j + 1) + (float)(n % 3 - 1);
        const float fx = floorf(px);
        const float fy = floorf(py);
        const int ltx = min(max((int)fx, 0), Hp - 1);
        const int lty = min(max((int)fy, 0), Wp - 1);
        const int rbx = min(max((int)fx + 1, 0), Hp - 1);
        const int rby = min(max((int)fy + 1, 0), Wp - 1);
        const float pxc = fminf(fmaxf(px, 0.0f), (float)(Hp - 1));
        const float pyc = fminf(fmaxf(py, 0.0f), (float)(Wp - 1));
        const float gx_lt = 1.0f + ((float)ltx - pxc);
        const float gy_lt = 1.0f + ((float)lty - pyc);
        const float gx_rb = 1.0f - ((float)rbx - pxc);
        const float gy_rb = 1.0f - ((float)rby - pyc);
        const float g_lt = gx_lt * gy_lt;
        const float g_rb = gx_rb * gy_rb;
        const float g_lb = gx_lt * gy_rb;
        const float g_rt = gx_rb * gy_lt;
        const int s_lt = ltx * Wp + lty;
        const int s_rb = rbx * Wp + rby;
        const int s_lb = ltx * Wp + rby;
        const int s_rt = rbx * Wp + lty;

        // --- 16 K-steps of 4 channels each; 4 WMMAs per K-step ---
        #pragma unroll 2
        for (int kb = 0; kb < 16; ++kb) {
            const int c0 = 4 * kb + 2 * hl;      // this lane's first channel
            const float* xc0 = xb + (size_t)c0 * HPWP;
            const float* xc1 = xc0 + HPWP;
            // A fragment: lanes 0-15 -> K0/K1, lanes 16-31 -> K2/K3
            const float a0 = g_lt * xc0[s_lt] + g_rb * xc0[s_rb] +
                             g_lb * xc0[s_lb] + g_rt * xc0[s_rt];
            const float a1 = g_lt * xc1[s_lt] + g_rb * xc1[s_rb] +
                             g_lb * xc1[s_lb] + g_rt * xc1[s_rt];
            v2f a = {a0, a1};
            // B fragments from LDS slab: row = channel, col = co
            const float* w0 = wchunk + (size_t)c0 * 64;
            const float* w1 = w0 + 64;
            v2f b0 = { w0[nid],      w1[nid]      };
            v2f b1 = { w0[nid + 16], w1[nid + 16] };
            v2f b2 = { w0[nid + 32], w1[nid + 32] };
            v2f b3 = { w0[nid + 48], w1[nid + 48] };
            acc0 = __builtin_amdgcn_wmma_f32_16x16x4_f32(false, a, false, b0, (short)0, acc0, false, false);
            acc1 = __builtin_amdgcn_wmma_f32_16x16x4_f32(false, a, false, b1, (short)0, acc1, false, false);
            acc2 = __builtin_amdgcn_wmma_f32_16x16x4_f32(false, a, false, b2, (short)0, acc2, false, false);
            acc3 = __builtin_amdgcn_wmma_f32_16x16x4_f32(false, a, false, b3, (short)0, acc3, false, false);
        }
    }

    // --- epilogue: C-tile layout -> [pixel][co] in padded LDS, coalesced store ---
    __syncthreads();
    float* os = &ostage[wave][0];
    #pragma unroll
    for (int r = 0; r < 8; ++r) {
        const int mm = r + 8 * hl;               // C layout: VGPR r holds M = r (+8 for hi half)
        os[mm * 65 + (nid +  0)] = acc0[r];
        os[mm * 65 + (nid + 16)] = acc1[r];
        os[mm * 65 + (nid + 32)] = acc2[r];
        os[mm * 65 + (nid + 48)] = acc3[r];
    }
    __syncthreads();
    // lanes 0-15 store co = 2*cw (16 consecutive pixels), lanes 16-31 co = 2*cw+1
    #pragma unroll 4
    for (int cw = 0; cw < 32; ++cw) {
        const int co = 2 * cw + hl;
        out[((size_t)(b * Csz + co)) * HW + pix0 + nid] = os[nid * 65 + co];
    }
}

// ---------------------------------------------------------------------------
// Launch
// ---------------------------------------------------------------------------
extern "C" void kernel_launch(void* const* d_in, const int* in_sizes, int n_in,
                              void* d_out, int out_size, void* d_ws, size_t ws_size,
                              hipStream_t stream) {
    const float* x        = (const float*)d_in[0];
    const float* p_conv_w = (const float*)d_in[1];
    const float* p_conv_b = (const float*)d_in[2];
    const float* conv_w   = (const float*)d_in[3];
    float* out = (float*)d_out;

    // workspace layout (floats): xpad | offsets | reordered weights
    const size_t n_xpad = (size_t)Bsz * Csz * HPWP;     // 8,652,800
    const size_t n_off  = (size_t)Bsz * 18 * HW;        // 2,359,296
    float* xpad = (float*)d_ws;
    float* offs = xpad + n_xpad;
    float* Wr   = offs + n_off;

    {
        int total = (int)n_xpad;
        pad_kernel<<<(total + 255) / 256, 256, 0, stream>>>(x, xpad, total);
    }
    {
        int total = NTAP * Csz * Csz;
        reorder_w_kernel<<<(total + 255) / 256, 256, 0, stream>>>(conv_w, Wr);
    }
    offset_conv_kernel<<<Bsz * 8 * 8, 256, 0, stream>>>(x, p_conv_w, p_conv_b, offs);
    deform_gemm_kernel<<<2048, 128, 0, stream>>>(xpad, offs, Wr, out);
}